// KTopAggregation_72086731096451
// MI455X (gfx1250) — compile-verified
//
#include <hip/hip_runtime.h>
#include <math.h>

// ---------------------------------------------------------------------------
// KTopAggregation on MI455X (gfx1250, wave32, WMMA + Tensor Data Mover)
//   K1: scores = leaky(x@W1+b1)@W2+b2          (f32 WMMA 16x16x4, TDM staging)
//   K2: per-graph softmax-max/sum + argmax pool (sorted batch -> binary search,
//       deterministic block tree reductions), writes feats = x[idx]*pmax
//   K3: out = leaky(feats@Wh+bh)               (f32 WMMA 16x16x4, TDM staging)
// ---------------------------------------------------------------------------

typedef __attribute__((ext_vector_type(2))) float v2f;
typedef __attribute__((ext_vector_type(8))) float v8f;
typedef __attribute__((ext_vector_type(4))) unsigned int v4u;
typedef __attribute__((ext_vector_type(4))) int v4i;
typedef __attribute__((ext_vector_type(8))) int v8i;

#define N_NODES 1000000
#define DIM     256
#define HID     64
#define KSC     3
#define NGRAPH  4096
#define NEG_SLOPE 0.01f

__device__ __forceinline__ float leaky(float v) { return v >= 0.0f ? v : NEG_SLOPE * v; }

__device__ __forceinline__ v8f wmma_f32_4(v2f a, v2f b, v8f c) {
  // D = A(16x4 f32) * B(4x16 f32) + C(16x16 f32)
  return __builtin_amdgcn_wmma_f32_16x16x4_f32(false, a, false, b, (short)0, c, false, false);
}

#if __has_builtin(__builtin_amdgcn_tensor_load_to_lds)
#define HAVE_TDM 1
#endif

#ifdef HAVE_TDM
__device__ __forceinline__ int rfl(unsigned int v) {
  return __builtin_amdgcn_readfirstlane((int)v);
}

// 2D tile load, 4-byte elements: tile (tile_d0 cols x tile_d1 rows) from a
// row-major tensor whose row stride is stride_d0 elements. Rows/cols beyond
// (tensor_d0, tensor_d1) — measured from the tile start — read as zero.
// LDS destination is filled contiguously row-major (tile_d0*4 bytes per row).
__device__ __forceinline__ void tdm_load_2d(unsigned int lds_off, const void* gaddr,
                                            unsigned int tile_d0, unsigned int tile_d1,
                                            unsigned int tensor_d0, unsigned int tensor_d1,
                                            unsigned int stride_d0) {
  unsigned long long ga = (unsigned long long)(uintptr_t)gaddr;
  v4u g0;
  g0[0] = (unsigned int)rfl(1u);                                   // count=1, user mode
  g0[1] = (unsigned int)rfl(lds_off);                              // lds_addr (bytes)
  g0[2] = (unsigned int)rfl((unsigned int)(ga & 0xFFFFFFFFu));     // global_addr[31:0]
  g0[3] = (unsigned int)rfl(((unsigned int)((ga >> 32) & 0x1FFFFFFu)) | (2u << 30)); // addr[56:32] | type=2

  v8i g1;
  g1[0] = rfl(2u << 16);                                           // wg_mask=0, data_size=4B
  g1[1] = rfl((tensor_d0 & 0xFFFFu) << 16);                        // abar=0 | tensor_dim0[15:0]
  g1[2] = rfl((tensor_d0 >> 16) | ((tensor_d1 & 0xFFFFu) << 16));  // td0[31:16] | td1[15:0]
  g1[3] = rfl((tensor_d1 >> 16) | ((tile_d0 & 0xFFFFu) << 16));    // td1[31:16] | tile_dim0
  g1[4] = rfl(tile_d1 & 0xFFFFu);                                  // tile_dim1 | tile_dim2=0
  g1[5] = rfl(stride_d0);                                          // tensor_dim0_stride[31:0]
  g1[6] = rfl(0u);                                                 // stride[47:32] | dim1_stride[15:0]
  g1[7] = rfl(0u);                                                 // dim1_stride[47:16]

  v4i z4 = {0, 0, 0, 0};
#if defined(__clang_major__) && (__clang_major__ >= 23)
  v8i z8 = {0, 0, 0, 0, 0, 0, 0, 0};
  __builtin_amdgcn_tensor_load_to_lds(g0, g1, z4, z4, z8, 0);
#else
  __builtin_amdgcn_tensor_load_to_lds(g0, g1, z4, z4, 0);
#endif
}

__device__ __forceinline__ unsigned int lds_offset(const void* p) {
  return (unsigned int)(uintptr_t)p;  // generic LDS address: low 32 bits = LDS byte offset
}
#endif  // HAVE_TDM

// ---------------------------------------------------------------------------
// Kernel 1: per-block 128 nodes (8 waves x 16 nodes), hidden = 64 cols.
// K=256 streamed in 4 chunks of 64. LDS: W1 chunk 16KB + per-wave A 32KB.
// ---------------------------------------------------------------------------
__global__ __launch_bounds__(256)
void scores_kernel(const float* __restrict__ x, const float* __restrict__ W1,
                   const float* __restrict__ b1, const float* __restrict__ W2,
                   const float* __restrict__ b2, float* __restrict__ scores) {
  __shared__ float sW1[64 * HID];      // 64 k-rows x 64 h-cols (16 KB)
  __shared__ float sBuf[8 * 16 * 64];  // per-wave A chunks; reused as hidden (32 KB)

  const int tid  = threadIdx.x;
  const int w    = tid >> 5;
  const int lane = tid & 31;
  const int half = lane >> 4;
  const int m    = lane & 15;
  const int nodeBase = blockIdx.x * 128;

  v8f acc[4];
#pragma unroll
  for (int t = 0; t < 4; ++t)
#pragma unroll
    for (int r = 0; r < 8; ++r) acc[t][r] = 0.0f;

  for (int c = 0; c < 4; ++c) {
    __syncthreads();
#ifdef HAVE_TDM
    {
      if (w == 0) {
        // W1 K-chunk: 64 rows x 64 cols, contiguous (stride 64)
        tdm_load_2d(lds_offset(sW1), W1 + (size_t)c * 64 * HID, 64, 64, 64, 64, 64);
      }
      // this wave's A tile: 16 nodes x 64 K, row stride 256 elements
      const int nb = nodeBase + w * 16;
      int rows = N_NODES - nb;
      rows = rows < 0 ? 0 : (rows > 16 ? 16 : rows);
      tdm_load_2d(lds_offset(sBuf) + (unsigned int)w * 4096u,
                  x + (size_t)nb * DIM + c * 64, 64, 16, 64, (unsigned int)rows, DIM);
      __builtin_amdgcn_s_wait_tensorcnt(0);
    }
#else
    {
      const float4* src = (const float4*)(W1 + (size_t)c * 64 * HID);
      float4*       dst = (float4*)sW1;
#pragma unroll
      for (int i = 0; i < 4; ++i) dst[i * 256 + tid] = src[i * 256 + tid];
      const int nb = nodeBase + w * 16;
#pragma unroll
      for (int i = 0; i < 8; ++i) {
        int q   = i * 32 + lane;
        int row = q >> 4;
        int cq  = q & 15;
        int node = nb + row;
        if (node >= N_NODES) node = N_NODES - 1;
        *(float4*)&sBuf[w * 1024 + row * 64 + cq * 4] =
            *(const float4*)&x[(size_t)node * DIM + c * 64 + cq * 4];
      }
      if (c < 3) {
        int node = nb + (lane >> 1);
        if (node >= N_NODES) node = N_NODES - 1;
        __builtin_prefetch(&x[(size_t)node * DIM + (c + 1) * 64 + (lane & 1) * 32], 0, 0);
      }
    }
#endif
    __syncthreads();
    // ---- 16 WMMA K-steps x 4 hidden tiles ----
    const float* As = &sBuf[w * 1024 + m * 64];
    for (int k4 = 0; k4 < 16; ++k4) {
      const int kb = k4 * 4 + 2 * half;   // lanes 0-15: K=k,k+1 ; 16-31: K=k+2,k+3
      v2f a = *(const v2f*)&As[kb];
#pragma unroll
      for (int t = 0; t < 4; ++t) {
        const int col = t * 16 + m;
        v2f b;
        b.x = sW1[kb * 64 + col];
        b.y = sW1[(kb + 1) * 64 + col];
        acc[t] = wmma_f32_4(a, b, acc[t]);
      }
    }
  }
  __syncthreads();

  // ---- epilogue: bias + leaky, spill hidden to LDS (reuse sBuf) ----
  float* hid = sBuf;  // [nodeLocal 0..127][h 0..63]
#pragma unroll
  for (int t = 0; t < 4; ++t) {
    const int col = t * 16 + m;
    const float bb = b1[col];
#pragma unroll
    for (int r = 0; r < 8; ++r) {
      float v = leaky(acc[t][r] + bb);            // C/D layout: M = r + 8*half
      hid[(w * 16 + r + 8 * half) * 64 + col] = v;
    }
  }
  __syncthreads();

  // ---- tiny H->K GEMM: one thread per node ----
  if (tid < 128) {
    const int node = nodeBase + tid;
    if (node < N_NODES) {
      float s0 = b2[0], s1 = b2[1], s2 = b2[2];
      const float* hrow = &hid[tid * 64];
#pragma unroll
      for (int h = 0; h < HID; ++h) {
        const float v = hrow[h];
        s0 += v * W2[h * 3 + 0];
        s1 += v * W2[h * 3 + 1];
        s2 += v * W2[h * 3 + 2];
      }
      scores[(size_t)node * 3 + 0] = s0;
      scores[(size_t)node * 3 + 1] = s1;
      scores[(size_t)node * 3 + 2] = s2;
    }
  }
}

// ---------------------------------------------------------------------------
// Kernel 2: one block per graph. batch is sorted -> binary search range, then
// deterministic tree reductions: max, sum(exp), (pmax, min node index).
// feats[g, k*256 + d] = x[idx_k, d] * pmax_k   (sc == pmax by construction)
// ---------------------------------------------------------------------------
__global__ __launch_bounds__(256)
void pool_kernel(const float* __restrict__ x, const float* __restrict__ scores,
                 const int* __restrict__ batch, float* __restrict__ feats) {
  const int g   = blockIdx.x;
  const int tid = threadIdx.x;

  int lo = 0, hi = N_NODES;
  while (lo < hi) { int mid = (lo + hi) >> 1; if (batch[mid] <  g)     lo = mid + 1; else hi = mid; }
  const int start = lo;
  lo = start; hi = N_NODES;
  while (lo < hi) { int mid = (lo + hi) >> 1; if (batch[mid] <  g + 1) lo = mid + 1; else hi = mid; }
  const int end = lo;

  __shared__ float red[3][256];
  __shared__ int   redi[3][256];
  __shared__ float mS[3], sS[3], pS[3];
  __shared__ int   iS[3];

  const float NEG_INF = -__builtin_inff();

  // ---- phase 1: segment max ----
  float lm[3] = {NEG_INF, NEG_INF, NEG_INF};
  for (int i = start + tid; i < end; i += 256) {
#pragma unroll
    for (int k = 0; k < 3; ++k) lm[k] = fmaxf(lm[k], scores[(size_t)i * 3 + k]);
  }
#pragma unroll
  for (int k = 0; k < 3; ++k) red[k][tid] = lm[k];
  __syncthreads();
  for (int off = 128; off > 0; off >>= 1) {
    if (tid < off)
#pragma unroll
      for (int k = 0; k < 3; ++k) red[k][tid] = fmaxf(red[k][tid], red[k][tid + off]);
    __syncthreads();
  }
  if (tid == 0)
#pragma unroll
    for (int k = 0; k < 3; ++k) mS[k] = red[k][0];
  __syncthreads();
  const float m0 = mS[0], m1 = mS[1], m2 = mS[2];

  // ---- phase 2: segment sum of exp ----
  float ls[3] = {0.0f, 0.0f, 0.0f};
  for (int i = start + tid; i < end; i += 256) {
    ls[0] += __expf(scores[(size_t)i * 3 + 0] - m0);
    ls[1] += __expf(scores[(size_t)i * 3 + 1] - m1);
    ls[2] += __expf(scores[(size_t)i * 3 + 2] - m2);
  }
#pragma unroll
  for (int k = 0; k < 3; ++k) red[k][tid] = ls[k];
  __syncthreads();
  for (int off = 128; off > 0; off >>= 1) {
    if (tid < off)
#pragma unroll
      for (int k = 0; k < 3; ++k) red[k][tid] += red[k][tid + off];
    __syncthreads();
  }
  if (tid == 0)
#pragma unroll
    for (int k = 0; k < 3; ++k) sS[k] = red[k][0];
  __syncthreads();
  const float s0 = sS[0], s1 = sS[1], s2 = sS[2];

  // ---- phase 3: pmax with min-node-id tie break ----
  float bp[3] = {NEG_INF, NEG_INF, NEG_INF};
  int   bi[3] = {N_NODES, N_NODES, N_NODES};
  for (int i = start + tid; i < end; i += 256) {
    float p0 = __expf(scores[(size_t)i * 3 + 0] - m0) / s0;
    float p1 = __expf(scores[(size_t)i * 3 + 1] - m1) / s1;
    float p2 = __expf(scores[(size_t)i * 3 + 2] - m2) / s2;
    if (p0 > bp[0]) { bp[0] = p0; bi[0] = i; }
    if (p1 > bp[1]) { bp[1] = p1; bi[1] = i; }
    if (p2 > bp[2]) { bp[2] = p2; bi[2] = i; }
  }
#pragma unroll
  for (int k = 0; k < 3; ++k) { red[k][tid] = bp[k]; redi[k][tid] = bi[k]; }
  __syncthreads();
  for (int off = 128; off > 0; off >>= 1) {
    if (tid < off) {
#pragma unroll
      for (int k = 0; k < 3; ++k) {
        const float p2v = red[k][tid + off];
        const int   i2v = redi[k][tid + off];
        if (p2v > red[k][tid]) { red[k][tid] = p2v; redi[k][tid] = i2v; }
        else if (p2v == red[k][tid] && i2v < redi[k][tid]) { redi[k][tid] = i2v; }
      }
    }
    __syncthreads();
  }
  if (tid == 0)
#pragma unroll
    for (int k = 0; k < 3; ++k) { pS[k] = red[k][0]; iS[k] = redi[k][0]; }
  __syncthreads();

  // ---- phase 4: gather + scale into feats (G x 768) ----
#pragma unroll
  for (int k = 0; k < 3; ++k) {
    int idx = iS[k];
    if (idx > N_NODES - 1) idx = N_NODES - 1;
    const float sc = pS[k];
    feats[(size_t)g * (KSC * DIM) + k * DIM + tid] = x[(size_t)idx * DIM + tid] * sc;
  }
}

// ---------------------------------------------------------------------------
// Kernel 3: out = leaky(feats @ Wh + bh). M=4096, K=768, N=256.
// Block tile 64x64, 8 waves -> each wave 16x32 (two 16x16 f32 WMMA tiles).
// K streamed in 12 chunks of 64. LDS: 16KB A + 16KB B, staged by the TDM.
// ---------------------------------------------------------------------------
__global__ __launch_bounds__(256)
void out_kernel(const float* __restrict__ feats, const float* __restrict__ Wh,
                const float* __restrict__ bh, float* __restrict__ out) {
  __shared__ float sF[64 * 64];
  __shared__ float sW[64 * 64];

  const int tid  = threadIdx.x;
  const int w    = tid >> 5;
  const int lane = tid & 31;
  const int half = lane >> 4;
  const int m    = lane & 15;
  const int blockRow = blockIdx.x * 64;
  const int blockCol = blockIdx.y * 64;
  const int wr = (w & 3) * 16;   // wave row offset within block tile
  const int wc = (w >> 2) * 32;  // wave col offset within block tile

  v8f acc[2];
#pragma unroll
  for (int t = 0; t < 2; ++t)
#pragma unroll
    for (int r = 0; r < 8; ++r) acc[t][r] = 0.0f;

  for (int c = 0; c < 12; ++c) {
    __syncthreads();
#ifdef HAVE_TDM
    if (w == 0) {
      // feats tile: 64 rows x 64 cols, row stride 768
      tdm_load_2d(lds_offset(sF), feats + (size_t)blockRow * (KSC * DIM) + c * 64,
                  64, 64, 64, 64, KSC * DIM);
      // Wh tile: 64 rows x 64 cols, row stride 256
      tdm_load_2d(lds_offset(sW), Wh + (size_t)(c * 64) * DIM + blockCol,
                  64, 64, 64, 64, DIM);
    }
    __builtin_amdgcn_s_wait_tensorcnt(0);
#else
#pragma unroll
    for (int i = 0; i < 4; ++i) {
      const int lin = i * 1024 + tid * 4;
      const int row = lin >> 6;
      const int col = lin & 63;
      *(float4*)&sF[lin] = *(const float4*)&feats[(size_t)(blockRow + row) * (KSC * DIM) + c * 64 + col];
      *(float4*)&sW[lin] = *(const float4*)&Wh[(size_t)(c * 64 + row) * DIM + blockCol + col];
    }
#endif
    __syncthreads();
    const float* As = &sF[(wr + m) * 64];
    for (int k4 = 0; k4 < 16; ++k4) {
      const int kb = k4 * 4 + 2 * half;
      v2f a = *(const v2f*)&As[kb];
#pragma unroll
      for (int t = 0; t < 2; ++t) {
        const int col = wc + t * 16 + m;
        v2f b;
        b.x = sW[kb * 64 + col];
        b.y = sW[(kb + 1) * 64 + col];
        acc[t] = wmma_f32_4(a, b, acc[t]);
      }
    }
  }

#pragma unroll
  for (int t = 0; t < 2; ++t) {
    const int col = blockCol + wc + t * 16 + m;
    const float bb = bh[col];
#pragma unroll
    for (int r = 0; r < 8; ++r) {
      const int row = blockRow + wr + r + 8 * half;
      out[(size_t)row * DIM + col] = leaky(acc[t][r] + bb);
    }
  }
}

// ---------------------------------------------------------------------------
extern "C" void kernel_launch(void* const* d_in, const int* in_sizes, int n_in,
                              void* d_out, int out_size, void* d_ws, size_t ws_size,
                              hipStream_t stream) {
  const float* x     = (const float*)d_in[0];
  const float* W1    = (const float*)d_in[1];
  const float* b1    = (const float*)d_in[2];
  const float* W2    = (const float*)d_in[3];
  const float* b2    = (const float*)d_in[4];
  const float* Wh    = (const float*)d_in[5];
  const float* bh    = (const float*)d_in[6];
  const int*   batch = (const int*)d_in[7];
  (void)in_sizes; (void)n_in; (void)out_size; (void)ws_size;

  float* scores = (float*)d_ws;                       // N x 3   (12.0 MB)
  float* feats  = scores + (size_t)N_NODES * KSC;     // G x 768 (12.6 MB)
  float* out    = (float*)d_out;                      // G x 256

  scores_kernel<<<(N_NODES + 127) / 128, 256, 0, stream>>>(x, W1, b1, W2, b2, scores);
  pool_kernel<<<NGRAPH, 256, 0, stream>>>(x, scores, batch, feats);
  out_kernel<<<dim3(NGRAPH / 64, DIM / 64), 256, 0, stream>>>(feats, Wh, bh, out);
}